// LightGCN_70300024701478
// MI455X (gfx1250) — compile-verified
//
#include <hip/hip_runtime.h>
#include <math.h>

typedef __attribute__((ext_vector_type(2))) float v2f;
typedef __attribute__((ext_vector_type(8))) float v8f;

#define N_USER   200000
#define N_ITEM   100000
#define NTOT     (N_USER + N_ITEM)
#define DIM      64
#define NNZ_E    1250000
#define BATCH    4096
#define NLAYERS  3

// ---------------------------------------------------------------------------
// 1) concat user/item embeddings into the dense [N, 64] table
// ---------------------------------------------------------------------------
__global__ void k_concat(const float* __restrict__ ue, const float* __restrict__ ie,
                         float* __restrict__ emb) {
  long long i = (long long)blockIdx.x * blockDim.x + threadIdx.x;
  const long long total   = (long long)NTOT * DIM;
  const long long userTot = (long long)N_USER * DIM;
  if (i >= total) return;
  emb[i] = (i < userTot) ? ue[i] : ie[i - userTot];
}

// ---------------------------------------------------------------------------
// 2) gather layer-0 rows at user/item indices into small accumulators
// ---------------------------------------------------------------------------
__global__ void k_gather_init(const float* __restrict__ emb,
                              const long long* __restrict__ users,
                              const long long* __restrict__ items,
                              float* __restrict__ Uacc, float* __restrict__ Iacc) {
  int i = blockIdx.x * blockDim.x + threadIdx.x;
  if (i >= BATCH * DIM) return;
  int b = i >> 6, d = i & 63;
  Uacc[i] = emb[(long long)users[b] * DIM + d];
  Iacc[i] = emb[((long long)N_USER + items[b]) * DIM + d];
}

__global__ void k_gather_add(const float* __restrict__ emb,
                             const long long* __restrict__ users,
                             const long long* __restrict__ items,
                             float* __restrict__ Uacc, float* __restrict__ Iacc) {
  int i = blockIdx.x * blockDim.x + threadIdx.x;
  if (i >= BATCH * DIM) return;
  int b = i >> 6, d = i & 63;
  Uacc[i] += emb[(long long)users[b] * DIM + d];
  Iacc[i] += emb[((long long)N_USER + items[b]) * DIM + d];
}

// ---------------------------------------------------------------------------
// 3) SpMM scatter: one thread per (edge, feature) -> global f32 atomic add.
//    emb table (76.8MB) is L2-resident (192MB L2), gathers are coalesced in d.
// ---------------------------------------------------------------------------
__global__ void k_spmm(const float* __restrict__ cur, float* __restrict__ nxt,
                       const float* __restrict__ val,
                       const int* __restrict__ row, const int* __restrict__ col) {
  long long t = (long long)blockIdx.x * blockDim.x + threadIdx.x;
  if (t >= (long long)NNZ_E * DIM) return;
  int e = (int)(t >> 6);
  int d = (int)(t & 63);
  int r = row[e];
  int c = col[e];
  float v = val[e];
  atomicAdd(&nxt[(long long)r * DIM + d], v * cur[(long long)c * DIM + d]);
}

// ---------------------------------------------------------------------------
// 4) scoring GEMM: out = sigmoid((Uacc . Iacc^T) / 16)   [4096 x 4096]
//    f32 WMMA 16x16x4, one 16x16 tile per wave, 16 chained wmma over K=64.
//    Fragment layouts per CDNA5 ISA 7.12.2 (wave32):
//      A: lanes 0-15 => M=lane, K pair {k,k+1}; lanes 16-31 => K pair {k+2,k+3}
//      B: same striping with N across lanes
//      C/D: vgpr j => row j + 8*(lane>=16), col lane%16
// ---------------------------------------------------------------------------
__global__ __launch_bounds__(128) void k_gemm_sig(const float* __restrict__ U,
                                                  const float* __restrict__ I,
                                                  float* __restrict__ out) {
  int wave = threadIdx.x >> 5;      // 4 waves / block
  int lane = threadIdx.x & 31;
  int hi   = lane >> 4;             // 0: K{0,1}, 1: K{2,3}
  int l16  = lane & 15;
  int mtile = blockIdx.y;                 // 0..255
  int ntile = blockIdx.x * 4 + wave;      // 0..255

  const v2f* Ur = (const v2f*)(U + ((long long)mtile * 16 + l16) * DIM);
  const v2f* Ir = (const v2f*)(I + ((long long)ntile * 16 + l16) * DIM);

  v8f c = {};
#pragma unroll
  for (int k = 0; k < DIM; k += 4) {
    v2f a = Ur[(k >> 1) + hi];
    v2f b = Ir[(k >> 1) + hi];
    // 8 args: (neg_a, A, neg_b, B, c_mod, C, reuse_a, reuse_b)
    c = __builtin_amdgcn_wmma_f32_16x16x4_f32(false, a, false, b,
                                              (short)0, c, false, false);
  }

  const float inv16 = 1.0f / 16.0f;   // (1/4)*(1/4) layer-mean scaling
#pragma unroll
  for (int j = 0; j < 8; ++j) {
    int r  = mtile * 16 + j + 8 * hi;
    int cc = ntile * 16 + l16;
    float x = c[j] * inv16;
    out[(long long)r * BATCH + cc] = 1.0f / (1.0f + __expf(-x));
  }
}

// ---------------------------------------------------------------------------
// launcher
// ---------------------------------------------------------------------------
extern "C" void kernel_launch(void* const* d_in, const int* in_sizes, int n_in,
                              void* d_out, int out_size, void* d_ws, size_t ws_size,
                              hipStream_t stream) {
  const float*     user_emb = (const float*)d_in[0];
  const float*     item_emb = (const float*)d_in[1];
  const float*     adj_val  = (const float*)d_in[2];
  const int*       adj_row  = (const int*)d_in[3];
  const int*       adj_col  = (const int*)d_in[4];
  const long long* users    = (const long long*)d_in[5];   // int64 in reference
  const long long* items    = (const long long*)d_in[6];   // int64 in reference
  float* out = (float*)d_out;

  const size_t embElems = (size_t)NTOT * DIM;   // 19.2M floats = 76.8MB
  float* embA = (float*)d_ws;
  float* embB = embA + embElems;
  float* Uacc = embB + embElems;                // 4096*64 floats
  float* Iacc = Uacc + (size_t)BATCH * DIM;

  // layer-0 table
  {
    long long total = (long long)embElems;
    unsigned grid = (unsigned)((total + 255) / 256);
    k_concat<<<grid, 256, 0, stream>>>(user_emb, item_emb, embA);
  }
  k_gather_init<<<(BATCH * DIM + 255) / 256, 256, 0, stream>>>(embA, users, items,
                                                               Uacc, Iacc);

  // 3 propagation layers (ping-pong), folding layer sums into Uacc/Iacc
  float* cur = embA;
  float* nxt = embB;
  for (int layer = 0; layer < NLAYERS; ++layer) {
    hipMemsetAsync(nxt, 0, embElems * sizeof(float), stream);
    long long total = (long long)NNZ_E * DIM;   // 80M (edge, feature) pairs
    unsigned grid = (unsigned)((total + 255) / 256);
    k_spmm<<<grid, 256, 0, stream>>>(cur, nxt, adj_val, adj_row, adj_col);
    k_gather_add<<<(BATCH * DIM + 255) / 256, 256, 0, stream>>>(nxt, users, items,
                                                                Uacc, Iacc);
    float* t = cur; cur = nxt; nxt = t;
  }

  // final WMMA scoring GEMM + sigmoid
  dim3 grid(BATCH / 16 / 4, BATCH / 16);  // (64, 256); 4 waves per block
  k_gemm_sig<<<grid, 128, 0, stream>>>(Uacc, Iacc, out);
}